// RCNN_87222195847717
// MI455X (gfx1250) — compile-verified
//
#include <hip/hip_runtime.h>
#include <math.h>

// ---- problem constants (match reference) ----
#define BDIM   4
#define NPROP  128
#define CCH    256
#define HF     23
#define WF     30
#define PH     7
#define PW     7
#define KH     5        // HF//PH + 2
#define KW     6        // WF//PW + 2
#define HID    256
#define OUTD   5
#define KF     (CCH*PH*PW)     // 12544
#define MROWS  (BDIM*NPROP)    // 512
#define NEGV   (-1e30f)
#define SCALE  (1.0f/32.0f)

#define KSPLIT 4
#define KSLICE (KF/KSPLIT)     // 3136

typedef float v2f __attribute__((ext_vector_type(2)));
typedef float v4f __attribute__((ext_vector_type(4)));
typedef float v8f __attribute__((ext_vector_type(8)));

// ---------------------------------------------------------------------------
// Kernel 1: quantized ROI max pooling.  One block per proposal (512 blocks),
// 256 threads cover the 12544 (c,ph,pw) output elements.
// Writes flat A matrix: (512 x 12544) row-major into workspace.
// ---------------------------------------------------------------------------
__global__ void roi_pool_kernel(const float* __restrict__ feat,
                                const float* __restrict__ prop,
                                float* __restrict__ flatA) {
  int bn = blockIdx.x;                // 0..511  (b*N + n)
  int b  = bn / NPROP;
  const float* box = prop + (size_t)bn * 5;
  float bx1 = box[0] * SCALE, by1 = box[1] * SCALE;
  float bx2 = box[2] * SCALE, by2 = box[3] * SCALE;
  int x1 = (int)floorf(bx1); x1 = x1 < 0 ? 0 : (x1 > WF - 1 ? WF - 1 : x1);
  int y1 = (int)floorf(by1); y1 = y1 < 0 ? 0 : (y1 > HF - 1 ? HF - 1 : y1);
  int rw = (int)ceilf(bx2) - x1; rw = rw < 1 ? 1 : (rw > WF ? WF : rw);
  int rh = (int)ceilf(by2) - y1; rh = rh < 1 ? 1 : (rh > HF ? HF : rh);

  const float* fb = feat + (size_t)b * CCH * HF * WF;
  for (int idx = threadIdx.x; idx < KF; idx += blockDim.x) {
    int c   = idx / (PH * PW);
    int rem = idx - c * (PH * PW);
    int ph  = rem / PW;
    int pw  = rem - ph * PW;
    int hs  = y1 + (ph * rh) / PH;
    int he  = y1 + ((ph + 1) * rh + PH - 1) / PH;
    int wsx = x1 + (pw * rw) / PW;
    int we  = x1 + ((pw + 1) * rw + PW - 1) / PW;
    const float* fc = fb + (size_t)c * HF * WF;
    float m = NEGV;
    #pragma unroll
    for (int j = 0; j < KH; ++j) {
      int hi = hs + j;
      if (hi < he && hi < HF) {
        const float* frow = fc + hi * WF;
        #pragma unroll
        for (int i = 0; i < KW; ++i) {
          int wi = wsx + i;
          if (wi < we && wi < WF) {
            float v = frow[wi];
            m = v > m ? v : m;
          }
        }
      }
    }
    flatA[(size_t)bn * KF + idx] = m;
  }
}

// ---------------------------------------------------------------------------
// Kernel 2: FC1 partial GEMM via V_WMMA_F32_16X16X4_F32.
// Each wave computes a 32x32 output tile over a K-slice of 3136:
//   - 2x2 subtiles of 16x16 -> 4 independent accumulator chains
//   - each A/B fragment reused by 2 WMMAs (halves L1/L2 traffic)
// Waves: 16 Mtiles * 8 Ntiles * 4 Kslices = 512 waves = 64 blocks * 8 waves.
// Partials written to ws; reduced (+bias,+ReLU) by fc1_reduce_kernel.
// A fragment (ISA 32-bit 16x4 layout): lanes 0-15 hold (M=lane, K=k..k+1),
// lanes 16-31 hold (M=lane-16, K=k+2..k+3)  -> contiguous float2 per lane.
// B assumed symmetric (N striped across lanes, same K split).
// ---------------------------------------------------------------------------
__global__ void gemm1_wmma_kernel(const float* __restrict__ A,
                                  const float* __restrict__ W1,
                                  float* __restrict__ partial) {
  int wave  = (int)((blockIdx.x * blockDim.x + threadIdx.x) >> 5);  // 0..511
  int lane  = threadIdx.x & 31;
  int ks    = wave & (KSPLIT - 1);      // K-slice id
  int tile  = wave >> 2;                // 0..127
  int tileM = tile & 15;                // 16 M-tiles of 32
  int tileN = tile >> 4;                // 8  N-tiles of 32
  int m0 = tileM * 32;
  int n0 = tileN * 32;
  int k0 = ks * KSLICE;
  int half = lane >> 4;
  int l15  = lane & 15;

  const float* pa0 = A  + (size_t)(m0 + l15)      * KF + k0 + half * 2;
  const float* pa1 = A  + (size_t)(m0 + 16 + l15) * KF + k0 + half * 2;
  const float* pb0 = W1 + (size_t)(n0 + l15)      * KF + k0 + half * 2;
  const float* pb1 = W1 + (size_t)(n0 + 16 + l15) * KF + k0 + half * 2;

  v8f c00 = {0,0,0,0,0,0,0,0};
  v8f c01 = {0,0,0,0,0,0,0,0};
  v8f c10 = {0,0,0,0,0,0,0,0};
  v8f c11 = {0,0,0,0,0,0,0,0};

  #pragma unroll 2
  for (int k = 0; k < KSLICE; k += 4) {
    v2f a0 = *(const v2f*)(pa0 + k);
    v2f a1 = *(const v2f*)(pa1 + k);
    v2f b0 = *(const v2f*)(pb0 + k);
    v2f b1 = *(const v2f*)(pb1 + k);
    c00 = __builtin_amdgcn_wmma_f32_16x16x4_f32(false, a0, false, b0,
                                                (short)0, c00, false, false);
    c01 = __builtin_amdgcn_wmma_f32_16x16x4_f32(false, a0, false, b1,
                                                (short)0, c01, false, false);
    c10 = __builtin_amdgcn_wmma_f32_16x16x4_f32(false, a1, false, b0,
                                                (short)0, c10, false, false);
    c11 = __builtin_amdgcn_wmma_f32_16x16x4_f32(false, a1, false, b1,
                                                (short)0, c11, false, false);
  }

  // C/D layout: VGPR r holds (M = r, N = lane) for lanes 0-15 and
  // (M = r+8, N = lane-16) for lanes 16-31.
  float* pbase = partial + (size_t)ks * MROWS * HID;
  int col0 = n0 + l15;
  #pragma unroll
  for (int r = 0; r < 8; ++r) {
    int row0 = m0 + r + half * 8;
    pbase[(size_t)row0        * HID + col0     ] = c00[r];
    pbase[(size_t)row0        * HID + col0 + 16] = c01[r];
    pbase[(size_t)(row0 + 16) * HID + col0     ] = c10[r];
    pbase[(size_t)(row0 + 16) * HID + col0 + 16] = c11[r];
  }
}

// ---------------------------------------------------------------------------
// Kernel 2b: reduce K-slice partials, add bias, ReLU -> H (512 x 256).
// Deterministic (fixed summation order) -- no float atomics.
// ---------------------------------------------------------------------------
__global__ void fc1_reduce_kernel(const float* __restrict__ partial,
                                  const float* __restrict__ b1,
                                  float* __restrict__ H) {
  int idx = blockIdx.x * blockDim.x + threadIdx.x;   // 0 .. 512*256-1
  if (idx >= MROWS * HID) return;
  int col = idx & (HID - 1);
  float s = b1[col];
  #pragma unroll
  for (int ks = 0; ks < KSPLIT; ++ks)
    s += partial[(size_t)ks * MROWS * HID + idx];
  H[idx] = s > 0.0f ? s : 0.0f;
}

// ---------------------------------------------------------------------------
// Kernel 3: FC2 (512x256 @ 256x5) + bias + box refinement + sigmoid(conf).
// One thread per (row, out) element: 2560 threads.  float4 dot product.
// ---------------------------------------------------------------------------
__global__ void head_kernel(const float* __restrict__ H,
                            const float* __restrict__ W2,
                            const float* __restrict__ b2,
                            const float* __restrict__ prop,
                            float* __restrict__ out) {
  int idx = blockIdx.x * blockDim.x + threadIdx.x;
  if (idx >= MROWS * OUTD) return;
  int m = idx / OUTD;
  int o = idx - m * OUTD;
  const v4f* hr = (const v4f*)(H  + (size_t)m * HID);
  const v4f* wr = (const v4f*)(W2 + (size_t)o * HID);
  float acc = b2[o];
  #pragma unroll 4
  for (int k = 0; k < HID / 4; ++k) {
    v4f h4 = hr[k];
    v4f w4 = wr[k];
    acc += h4.x * w4.x + h4.y * w4.y + h4.z * w4.z + h4.w * w4.w;
  }
  float p = prop[(size_t)m * 5 + o];
  float res;
  if (o < 4) {
    res = p + acc;
  } else {
    res = 1.0f / (1.0f + __expf(-(p + acc)));
  }
  out[idx] = res;
}

// ---------------------------------------------------------------------------
extern "C" void kernel_launch(void* const* d_in, const int* in_sizes, int n_in,
                              void* d_out, int out_size, void* d_ws, size_t ws_size,
                              hipStream_t stream) {
  const float* feat = (const float*)d_in[0];   // (B,C,HF,WF)
  const float* prop = (const float*)d_in[1];   // (B,N,5)
  const float* W1   = (const float*)d_in[2];   // (HID, 12544)
  const float* b1   = (const float*)d_in[3];   // (HID,)
  const float* W2   = (const float*)d_in[4];   // (OUT, HID)
  const float* b2   = (const float*)d_in[5];   // (OUT,)

  char*  ws      = (char*)d_ws;
  float* flatA   = (float*)ws;                                    // 512 x 12544
  size_t offA    = (size_t)MROWS * KF * sizeof(float);            // 25.69 MB
  float* partial = (float*)(ws + offA);                           // 4 x 512 x 256
  size_t offP    = offA + (size_t)KSPLIT * MROWS * HID * sizeof(float);
  float* H       = (float*)(ws + offP);                           // 512 x 256

  roi_pool_kernel<<<MROWS, 256, 0, stream>>>(feat, prop, flatA);

  // 512 tile-waves / 8 waves per 256-thread block = 64 blocks
  gemm1_wmma_kernel<<<64, 256, 0, stream>>>(flatA, W1, partial);

  fc1_reduce_kernel<<<(MROWS * HID + 255) / 256, 256, 0, stream>>>(
      partial, b1, H);

  head_kernel<<<(MROWS * OUTD + 255) / 256, 256, 0, stream>>>(
      H, W2, b2, prop, (float*)d_out);
}